// FocalLossWithDice_75900662055022
// MI455X (gfx1250) — compile-verified
//
#include <hip/hip_runtime.h>

// ---------------------------------------------------------------------------
// FocalLossWithDice for MI455X (gfx1250, wave32)
//
// outputs: [16, 8, 512, 512] f32   targets: [16, 512, 512] int
// out:     scalar f32
//
// Pass 1 (bandwidth-bound, ~150 MB @ 23.3 TB/s ≈ 6.5 us floor):
//   streaming softmax/focal/dice-stats, b128 loads, register accumulators,
//   wave shuffle + LDS reduction, deterministic per-block partials in d_ws.
// Pass 2 (one wave32):
//   reduce block partials; dice table [16x8] reduced with two chained
//   V_WMMA_F32_16X16X4_F32 (full-f32 matrix op: D = A_lo*1 + (A_hi*1 + 0)).
// ---------------------------------------------------------------------------

#define NUM_CLASSES     8
#define IGNORE_INDEX    255
#define NPIX            (512 * 512)          // pixels per image
#define BATCH           16
#define EPSV            1e-5f
#define MIN_CLASS_PIX   10.0f
#define CE_W            1.0f
#define D_W             0.1f

#define BLOCKS_PER_IMG  32
#define THREADS         256
#define TOTAL_BLOCKS    (BATCH * BLOCKS_PER_IMG)   // 512
#define WS_STRIDE       32                         // floats per block record
// block record layout: [0..7]=inter, [8..15]=psum, [16..23]=tsum, [24]=ce, [25]=cnt

typedef __attribute__((ext_vector_type(2))) float v2f;
typedef __attribute__((ext_vector_type(8))) float v8f;

__device__ __forceinline__ float wave_red(float v) {
#pragma unroll
  for (int off = 16; off > 0; off >>= 1) v += __shfl_down(v, off, 32);
  return v;
}

__global__ __launch_bounds__(THREADS) void focal_dice_main(
    const float* __restrict__ outputs, const int* __restrict__ targets,
    float* __restrict__ ws) {
  const int b = blockIdx.y;
  const float* out_b = outputs + (size_t)b * ((size_t)NUM_CLASSES * NPIX);
  const int*   tgt_b = targets + (size_t)b * NPIX;

  float inter[8], psum[8], tsum[8];
#pragma unroll
  for (int c = 0; c < 8; ++c) { inter[c] = 0.f; psum[c] = 0.f; tsum[c] = 0.f; }
  float ce = 0.f, cnt = 0.f;

  const int tid     = blockIdx.x * THREADS + threadIdx.x;
  const int nthr    = BLOCKS_PER_IMG * THREADS;
  const int ngroups = NPIX / 4;

  for (int g = tid; g < ngroups; g += nthr) {
    const int p = g * 4;
    const int4 t4 = *reinterpret_cast<const int4*>(tgt_b + p);
    float4 x4[8];
#pragma unroll
    for (int c = 0; c < 8; ++c)
      x4[c] = *reinterpret_cast<const float4*>(out_b + (size_t)c * NPIX + p);
    const int tj[4] = { t4.x, t4.y, t4.z, t4.w };

#pragma unroll
    for (int j = 0; j < 4; ++j) {
      float xv[8];
#pragma unroll
      for (int c = 0; c < 8; ++c)
        xv[c] = reinterpret_cast<const float*>(&x4[c])[j];

      const int   t     = tj[j];
      const bool  valid = (t != IGNORE_INDEX);
      const int   ts    = valid ? t : 0;
      const float vf    = valid ? 1.0f : 0.0f;

      float m = xv[0];
#pragma unroll
      for (int c = 1; c < 8; ++c) m = fmaxf(m, xv[c]);
      float e[8], s = 0.f;
#pragma unroll
      for (int c = 0; c < 8; ++c) { e[c] = __expf(xv[c] - m); s += e[c]; }
      const float inv  = 1.0f / s;
      const float logs = __logf(s);

#pragma unroll
      for (int c = 0; c < 8; ++c) {
        const float pr  = e[c] * inv;
        psum[c] += pr * vf;
        const float sel = (c == ts) ? vf : 0.0f;
        const float lp  = xv[c] - m - logs;       // log-softmax of channel c
        const float om  = 1.0f - pr;
        ce       += -(om * om * lp) * sel;        // focal NLL at target channel
        inter[c] += pr * sel;
        tsum[c]  += sel;
      }
      cnt += vf;
    }
  }

  // ---- intra-block reduction: shuffle within wave, then LDS across 8 waves
  __shared__ float s_part[THREADS / 32][26];
  const int wave = threadIdx.x >> 5;
  const int lane = threadIdx.x & 31;
#pragma unroll
  for (int c = 0; c < 8; ++c) {
    inter[c] = wave_red(inter[c]);
    psum[c]  = wave_red(psum[c]);
    tsum[c]  = wave_red(tsum[c]);
  }
  ce  = wave_red(ce);
  cnt = wave_red(cnt);
  if (lane == 0) {
#pragma unroll
    for (int c = 0; c < 8; ++c) {
      s_part[wave][c]      = inter[c];
      s_part[wave][8 + c]  = psum[c];
      s_part[wave][16 + c] = tsum[c];
    }
    s_part[wave][24] = ce;
    s_part[wave][25] = cnt;
  }
  __syncthreads();
  if (threadIdx.x < 26) {
    float acc = 0.f;
#pragma unroll
    for (int w = 0; w < THREADS / 32; ++w) acc += s_part[w][threadIdx.x];
    ws[(size_t)(b * BLOCKS_PER_IMG + blockIdx.x) * WS_STRIDE + threadIdx.x] = acc;
  }
}

// weighted, masked dice term for (image b, class c); D_W and 1/((C-1)*B) folded in
__device__ __forceinline__ float dice_term(const float* __restrict__ ws, int b, int c) {
  float it = 0.f, ps = 0.f, ts = 0.f;
  for (int g = 0; g < BLOCKS_PER_IMG; ++g) {
    const float* base = ws + (size_t)(b * BLOCKS_PER_IMG + g) * WS_STRIDE;
    it += base[c];
    ps += base[8 + c];
    ts += base[16 + c];
  }
  float per = 1.0f - (2.0f * it + EPSV) / (ps + ts + EPSV);
  per = (ts > MIN_CLASS_PIX) ? per : 0.0f;
  const float w = (c >= 1) ? (D_W / ((NUM_CLASSES - 1) * (float)BATCH)) : 0.0f;
  return per * w;
}

__global__ __launch_bounds__(32) void focal_dice_combine(
    const float* __restrict__ ws, float* __restrict__ out) {
  const int lane = threadIdx.x;   // one wave32, EXEC all ones at the WMMAs
  const int m  = lane & 15;       // A-matrix row (image index)
  const int kh = lane >> 4;       // K half per documented 16x4 f32 A layout

  // A_lo = per_cls[:, 0:4], A_hi = per_cls[:, 4:8]; lane holds A[m, 2*kh + j]
  v2f aLo, aHi, bOne;
  bOne[0] = 1.0f; bOne[1] = 1.0f;              // B = ones(4x16): layout-robust
#pragma unroll
  for (int j = 0; j < 2; ++j) {
    const int k = kh * 2 + j;
    aLo[j] = dice_term(ws, m, k);
    aHi[j] = dice_term(ws, m, k + 4);
  }

  // Two chained full-f32 WMMAs: D[m,n] = sum_{c=0..7} per_cls_weighted[m,c]
  v8f acc = {0.f, 0.f, 0.f, 0.f, 0.f, 0.f, 0.f, 0.f};
  acc = __builtin_amdgcn_wmma_f32_16x16x4_f32(false, aHi, false, bOne,
                                              (short)0, acc, false, false);
  acc = __builtin_amdgcn_wmma_f32_16x16x4_f32(false, aLo, false, bOne,
                                              (short)0, acc, false, false);

  // Column 0 of D: lane 0 holds M=0..7 in its 8 C/D VGPRs, lane 16 holds M=8..15
  float s = 0.f;
#pragma unroll
  for (int i = 0; i < 8; ++i) s += acc[i];
  const float d_total = __shfl(s, 0, 32) + __shfl(s, 16, 32);

  // CE reduction across all 512 block records, spread over the wave
  float ce_s = 0.f, cnt_s = 0.f;
  for (int blk = lane; blk < TOTAL_BLOCKS; blk += 32) {
    const float* base = ws + (size_t)blk * WS_STRIDE;
    ce_s  += base[24];
    cnt_s += base[25];
  }
  ce_s  = wave_red(ce_s);
  cnt_s = wave_red(cnt_s);

  if (lane == 0) out[0] = CE_W * (ce_s / cnt_s) + d_total;
}

extern "C" void kernel_launch(void* const* d_in, const int* in_sizes, int n_in,
                              void* d_out, int out_size, void* d_ws, size_t ws_size,
                              hipStream_t stream) {
  (void)in_sizes; (void)n_in; (void)out_size; (void)ws_size;
  const float* outputs = (const float*)d_in[0];
  const int*   targets = (const int*)d_in[1];
  float* out = (float*)d_out;
  float* ws  = (float*)d_ws;   // 512 blocks * 32 f32 = 64 KB used

  dim3 grid(BLOCKS_PER_IMG, BATCH);
  focal_dice_main<<<grid, THREADS, 0, stream>>>(outputs, targets, ws);
  focal_dice_combine<<<1, 32, 0, stream>>>(ws, out);
}